// VQVAEQuantizer_32238024524332
// MI455X (gfx1250) — compile-verified
//
#include <hip/hip_runtime.h>
#include <hip/hip_bf16.h>

typedef __attribute__((ext_vector_type(2))) float v2f;
typedef __attribute__((ext_vector_type(8))) float v8f;
typedef int v4i __attribute__((vector_size(16)));   // matches builtin param type

static constexpr int N_ROWS  = 16384;
static constexpr int IN_DIM  = 2048;
static constexpr int DIM     = 1024;
static constexpr int CB_SIZE = 8192;
static constexpr int TPB     = 256;

#define VQ_AS1 __attribute__((address_space(1)))
#define VQ_AS3 __attribute__((address_space(3)))

#if defined(__HIP_DEVICE_COMPILE__) && __has_builtin(__builtin_amdgcn_global_load_async_to_lds_b128)
#define VQ_HAVE_ASYNC 1
#else
#define VQ_HAVE_ASYNC 0
#endif

// 16-byte global -> LDS copy (async on CDNA5, sync fallback otherwise)
__device__ __forceinline__ void vq_cp16(float* lds_dst, const float* gsrc)
{
#if VQ_HAVE_ASYNC
    __builtin_amdgcn_global_load_async_to_lds_b128(
        (VQ_AS1 v4i*)gsrc, (VQ_AS3 v4i*)lds_dst, 0, 0);
#else
    *reinterpret_cast<float4*>(lds_dst) = *reinterpret_cast<const float4*>(gsrc);
#endif
}

template<int N>
__device__ __forceinline__ void vq_wait_async()
{
#if VQ_HAVE_ASYNC
#if __has_builtin(__builtin_amdgcn_s_wait_asynccnt)
    __builtin_amdgcn_s_wait_asynccnt(N);
#else
    asm volatile("s_wait_asynccnt %0" :: "i"(N) : "memory");
#endif
#endif
}

// =======================================================================
// GEMM1 (clip fused): C = clip(A) @ B + bias.  VGPR-staged (clip needs it).
// A:[M,K] B:[K,N] row-major. M%128==0, N%128==0, K%16==0.
// =======================================================================
__global__ __launch_bounds__(TPB)
void vq_gemm_clip(const float* __restrict__ A, const float* __restrict__ B,
                  const float* __restrict__ bias, float* __restrict__ C,
                  int M, int N, int K)
{
    constexpr int BM = 128, BN = 128, BK = 16, BKP = 20;
    __shared__ float As[BM * BKP];
    __shared__ float Bs[BK * BN];

    const int tid  = threadIdx.x;
    const int bm   = blockIdx.y * BM;
    const int bn   = blockIdx.x * BN;
    const int wave = tid >> 5;
    const int lane = tid & 31;
    const int wm   = (wave & 1) * 64;
    const int wn   = (wave >> 1) * 32;
    const int m16  = lane & 15;
    const int hi   = lane >> 4;
    const int kb2  = hi * 2;

    v8f acc[4][2];
    #pragma unroll
    for (int mt = 0; mt < 4; ++mt)
        #pragma unroll
        for (int nt = 0; nt < 2; ++nt)
            acc[mt][nt] = {};

    for (int kt = 0; kt < K; kt += BK) {
        #pragma unroll
        for (int i = 0; i < 2; ++i) {
            int idx = tid + i * TPB;
            int row = idx >> 2;
            int c4  = (idx & 3) << 2;
            float4 v = *reinterpret_cast<const float4*>(
                &A[(size_t)(bm + row) * K + kt + c4]);
            v.x = fminf(fmaxf(v.x, -10.f), 10.f);
            v.y = fminf(fmaxf(v.y, -10.f), 10.f);
            v.z = fminf(fmaxf(v.z, -10.f), 10.f);
            v.w = fminf(fmaxf(v.w, -10.f), 10.f);
            As[row * BKP + c4 + 0] = v.x;
            As[row * BKP + c4 + 1] = v.y;
            As[row * BKP + c4 + 2] = v.z;
            As[row * BKP + c4 + 3] = v.w;
        }
        #pragma unroll
        for (int i = 0; i < 2; ++i) {
            int idx = tid + i * TPB;
            int row = idx >> 5;
            int c4  = (idx & 31) << 2;
            float4 v = *reinterpret_cast<const float4*>(
                &B[(size_t)(kt + row) * N + bn + c4]);
            *reinterpret_cast<float4*>(&Bs[row * BN + c4]) = v;
        }
        __syncthreads();

        #pragma unroll
        for (int k0 = 0; k0 < BK; k0 += 4) {
            v2f afr[4], bfr[2];
            #pragma unroll
            for (int mt = 0; mt < 4; ++mt) {
                int r = wm + mt * 16 + m16;
                afr[mt].x = As[r * BKP + k0 + kb2 + 0];
                afr[mt].y = As[r * BKP + k0 + kb2 + 1];
            }
            #pragma unroll
            for (int nt = 0; nt < 2; ++nt) {
                int c = wn + nt * 16 + m16;
                bfr[nt].x = Bs[(k0 + kb2 + 0) * BN + c];
                bfr[nt].y = Bs[(k0 + kb2 + 1) * BN + c];
            }
            #pragma unroll
            for (int mt = 0; mt < 4; ++mt)
                #pragma unroll
                for (int nt = 0; nt < 2; ++nt)
                    acc[mt][nt] = __builtin_amdgcn_wmma_f32_16x16x4_f32(
                        false, afr[mt], false, bfr[nt],
                        (short)0, acc[mt][nt], false, false);
        }
        __syncthreads();
    }

    #pragma unroll
    for (int mt = 0; mt < 4; ++mt)
        #pragma unroll
        for (int nt = 0; nt < 2; ++nt) {
            int col = bn + wn + nt * 16 + m16;
            float b = bias[col];
            #pragma unroll
            for (int r = 0; r < 8; ++r) {
                int row = bm + wm + mt * 16 + hi * 8 + r;
                C[(size_t)row * N + col] = acc[mt][nt][r] + b;
            }
        }
}

// =======================================================================
// Async double-buffered GEMM: C = A @ B + bias (no clip).
// Tile t+1 streams global->LDS via ASYNC while tile t runs WMMA.
// =======================================================================
__global__ __launch_bounds__(TPB)
void vq_gemm_async(const float* __restrict__ A, const float* __restrict__ B,
                   const float* __restrict__ bias, float* __restrict__ C,
                   int M, int N, int K)
{
    constexpr int BM = 128, BN = 128, BK = 16, BKP = 20;
    __shared__ float As[2][BM * BKP];
    __shared__ float Bs[2][BK * BN];

    const int tid  = threadIdx.x;
    const int bm   = blockIdx.y * BM;
    const int bn   = blockIdx.x * BN;
    const int wave = tid >> 5;
    const int lane = tid & 31;
    const int wm   = (wave & 1) * 64;
    const int wn   = (wave >> 1) * 32;
    const int m16  = lane & 15;
    const int hi   = lane >> 4;
    const int kb2  = hi * 2;

    // per-thread staging coordinates (2 x 16B chunks per tile each for A and B)
    const int aRow0 = tid >> 2,            aC4 = (tid & 3) << 2;
    const int aRow1 = (tid + TPB) >> 2;
    const int bRow0 = tid >> 5,            bC4 = (tid & 31) << 2;
    const int bRow1 = (tid + TPB) >> 5;

    auto stage = [&](int buf, int kt) {
        vq_cp16(&As[buf][aRow0 * BKP + aC4], &A[(size_t)(bm + aRow0) * K + kt + aC4]);
        vq_cp16(&As[buf][aRow1 * BKP + aC4], &A[(size_t)(bm + aRow1) * K + kt + aC4]);
        vq_cp16(&Bs[buf][bRow0 * BN + bC4],  &B[(size_t)(kt + bRow0) * N + bn + bC4]);
        vq_cp16(&Bs[buf][bRow1 * BN + bC4],  &B[(size_t)(kt + bRow1) * N + bn + bC4]);
    };

    v8f acc[4][2];
    #pragma unroll
    for (int mt = 0; mt < 4; ++mt)
        #pragma unroll
        for (int nt = 0; nt < 2; ++nt)
            acc[mt][nt] = {};

    const int nK = K / BK;
    stage(0, 0);                       // 4 async ops outstanding
    for (int t = 0; t < nK; ++t) {
        const int buf = t & 1;
        if (t + 1 < nK) { stage(buf ^ 1, (t + 1) * BK); vq_wait_async<4>(); }
        else            { vq_wait_async<0>(); }
        __syncthreads();               // tile t visible to all waves

        const float* Ab = As[buf];
        const float* Bb = Bs[buf];
        #pragma unroll
        for (int k0 = 0; k0 < BK; k0 += 4) {
            v2f afr[4], bfr[2];
            #pragma unroll
            for (int mt = 0; mt < 4; ++mt) {
                int r = wm + mt * 16 + m16;
                afr[mt].x = Ab[r * BKP + k0 + kb2 + 0];
                afr[mt].y = Ab[r * BKP + k0 + kb2 + 1];
            }
            #pragma unroll
            for (int nt = 0; nt < 2; ++nt) {
                int c = wn + nt * 16 + m16;
                bfr[nt].x = Bb[(k0 + kb2 + 0) * BN + c];
                bfr[nt].y = Bb[(k0 + kb2 + 1) * BN + c];
            }
            #pragma unroll
            for (int mt = 0; mt < 4; ++mt)
                #pragma unroll
                for (int nt = 0; nt < 2; ++nt)
                    acc[mt][nt] = __builtin_amdgcn_wmma_f32_16x16x4_f32(
                        false, afr[mt], false, bfr[nt],
                        (short)0, acc[mt][nt], false, false);
        }
        __syncthreads();               // done reading buf before it is re-staged
    }

    #pragma unroll
    for (int mt = 0; mt < 4; ++mt)
        #pragma unroll
        for (int nt = 0; nt < 2; ++nt) {
            int col = bn + wn + nt * 16 + m16;
            float b = bias[col];
            #pragma unroll
            for (int r = 0; r < 8; ++r) {
                int row = bm + wm + mt * 16 + hi * 8 + r;
                C[(size_t)row * N + col] = acc[mt][nt][r] + b;
            }
        }
}

// =======================================================================
// Distance GEMM vs codebook ([CB,K] row-major) + fused per-tile argmin.
// Async double-buffered; both tiles stored [row][kpad] (b128-contiguous).
// d = ||c||^2 - 2*z.c  (row-constant ||z||^2 dropped: argmin-invariant)
// =======================================================================
__global__ __launch_bounds__(TPB)
void vq_dist_partial(const float* __restrict__ Ze, const float* __restrict__ Cb,
                     const float* __restrict__ cn,
                     float* __restrict__ pval, int* __restrict__ pidx,
                     int M, int K)
{
    constexpr int BM = 128, BK = 16, BKP = 20;
    __shared__ float As[2][BM * BKP];   // z_e rows
    __shared__ float Bs[2][BM * BKP];   // codebook rows (row-major, k contiguous)

    const int tid  = threadIdx.x;
    const int bm   = blockIdx.y * BM;
    const int bn   = blockIdx.x * BM;   // code block (128 codes)
    const int wave = tid >> 5;
    const int lane = tid & 31;
    const int wm   = (wave & 1) * 64;
    const int wn   = (wave >> 1) * 32;
    const int m16  = lane & 15;
    const int hi   = lane >> 4;
    const int kb2  = hi * 2;

    const int sRow0 = tid >> 2, sC4 = (tid & 3) << 2;
    const int sRow1 = (tid + TPB) >> 2;

    auto stage = [&](int buf, int kt) {
        vq_cp16(&As[buf][sRow0 * BKP + sC4], &Ze[(size_t)(bm + sRow0) * K + kt + sC4]);
        vq_cp16(&As[buf][sRow1 * BKP + sC4], &Ze[(size_t)(bm + sRow1) * K + kt + sC4]);
        vq_cp16(&Bs[buf][sRow0 * BKP + sC4], &Cb[(size_t)(bn + sRow0) * K + kt + sC4]);
        vq_cp16(&Bs[buf][sRow1 * BKP + sC4], &Cb[(size_t)(bn + sRow1) * K + kt + sC4]);
    };

    v8f acc[4][2];
    #pragma unroll
    for (int mt = 0; mt < 4; ++mt)
        #pragma unroll
        for (int nt = 0; nt < 2; ++nt)
            acc[mt][nt] = {};

    const int nK = K / BK;
    stage(0, 0);
    for (int t = 0; t < nK; ++t) {
        const int buf = t & 1;
        if (t + 1 < nK) { stage(buf ^ 1, (t + 1) * BK); vq_wait_async<4>(); }
        else            { vq_wait_async<0>(); }
        __syncthreads();

        const float* Ab = As[buf];
        const float* Bb = Bs[buf];
        #pragma unroll
        for (int k0 = 0; k0 < BK; k0 += 4) {
            v2f afr[4], bfr[2];
            #pragma unroll
            for (int mt = 0; mt < 4; ++mt) {
                int r = wm + mt * 16 + m16;
                afr[mt].x = Ab[r * BKP + k0 + kb2 + 0];
                afr[mt].y = Ab[r * BKP + k0 + kb2 + 1];
            }
            #pragma unroll
            for (int nt = 0; nt < 2; ++nt) {
                int c = wn + nt * 16 + m16;      // code row inside block
                bfr[nt].x = Bb[c * BKP + k0 + kb2 + 0];
                bfr[nt].y = Bb[c * BKP + k0 + kb2 + 1];
            }
            #pragma unroll
            for (int mt = 0; mt < 4; ++mt)
                #pragma unroll
                for (int nt = 0; nt < 2; ++nt)
                    acc[mt][nt] = __builtin_amdgcn_wmma_f32_16x16x4_f32(
                        false, afr[mt], false, bfr[nt],
                        (short)0, acc[mt][nt], false, false);
        }
        __syncthreads();
    }

    const int col0 = bn + wn + m16;
    const int col1 = col0 + 16;
    const float cn0 = cn[col0];
    const float cn1 = cn[col1];

    #pragma unroll
    for (int mt = 0; mt < 4; ++mt) {
        #pragma unroll
        for (int r = 0; r < 8; ++r) {
            float d0 = cn0 - 2.0f * acc[mt][0][r];
            float d1 = cn1 - 2.0f * acc[mt][1][r];
            float v  = d0; int id = col0;
            if (d1 < v) { v = d1; id = col1; }
            #pragma unroll
            for (int off = 1; off < 16; off <<= 1) {   // halves stay disjoint
                float ov = __shfl_xor(v, off, 32);
                int   oi = __shfl_xor(id, off, 32);
                if (ov < v || (ov == v && oi < id)) { v = ov; id = oi; }
            }
            if (m16 == 0) {
                int row = bm + wm + mt * 16 + hi * 8 + r;
                pval[(size_t)blockIdx.x * M + row] = v;
                pidx[(size_t)blockIdx.x * M + row] = id;
            }
        }
    }
}

// ---------------- small helper kernels ----------------
__global__ void vq_init(double* acc2, float* usage)
{
    int i = blockIdx.x * blockDim.x + threadIdx.x;
    if (i < 2) acc2[i] = 0.0;
    if (i < CB_SIZE) usage[i] = 0.0f;
}

__global__ __launch_bounds__(TPB)
void vq_rowsumsq(const float* __restrict__ X, float* __restrict__ out, int K)
{
    int wave = threadIdx.x >> 5, lane = threadIdx.x & 31;
    int row  = blockIdx.x * 8 + wave;
    const float4* p = reinterpret_cast<const float4*>(X + (size_t)row * K);
    float s = 0.f;
    for (int i = lane; i < K / 4; i += 32) {
        float4 v = p[i];
        s += v.x * v.x + v.y * v.y + v.z * v.z + v.w * v.w;
    }
    #pragma unroll
    for (int off = 16; off; off >>= 1) s += __shfl_xor(s, off, 32);
    if (lane == 0) out[row] = s;
}

__global__ void vq_argmin_reduce(const float* __restrict__ pval,
                                 const int* __restrict__ pidx,
                                 int nParts, int M,
                                 int* __restrict__ ids_i, float* __restrict__ ids_f)
{
    int row = blockIdx.x * blockDim.x + threadIdx.x;
    if (row >= M) return;
    float best = pval[row];
    int   bid  = pidx[row];
    for (int p = 1; p < nParts; ++p) {
        float v = pval[(size_t)p * M + row];
        int   i = pidx[(size_t)p * M + row];
        if (v < best || (v == best && i < bid)) { best = v; bid = i; }
    }
    ids_i[row] = bid;
    ids_f[row] = (float)bid;
}

__global__ void vq_hist(const int* __restrict__ ids, float* __restrict__ usage, int n)
{
    int i = blockIdx.x * blockDim.x + threadIdx.x;
    if (i < n) atomicAdd(&usage[ids[i]], 1.0f);
}

__device__ inline float vq_block_reduce(float s)
{
    __shared__ float warp[8];
    #pragma unroll
    for (int off = 16; off; off >>= 1) s += __shfl_xor(s, off, 32);
    int wave = threadIdx.x >> 5, lane = threadIdx.x & 31;
    if (lane == 0) warp[wave] = s;
    __syncthreads();
    if (wave == 0) {
        s = (lane < 8) ? warp[lane] : 0.f;
        #pragma unroll
        for (int off = 4; off; off >>= 1) s += __shfl_xor(s, off, 32);
    }
    return s;
}

__global__ __launch_bounds__(TPB)
void vq_gather_embloss(const int* __restrict__ ids, const float* __restrict__ cb,
                       const float* __restrict__ ze, float* __restrict__ zq,
                       double* __restrict__ acc)
{
    int row = blockIdx.x;
    int id  = ids[row];
    const float4* src  = reinterpret_cast<const float4*>(cb + (size_t)id  * DIM);
    const float4* zrow = reinterpret_cast<const float4*>(ze + (size_t)row * DIM);
    float4*       dst  = reinterpret_cast<float4*>(zq + (size_t)row * DIM);
    float s = 0.f;
    for (int i = threadIdx.x; i < DIM / 4; i += TPB) {
        float4 q = src[i], e = zrow[i];
        dst[i] = q;
        float dx = q.x - e.x, dy = q.y - e.y, dz = q.z - e.z, dw = q.w - e.w;
        s += dx * dx + dy * dy + dz * dz + dw * dw;
    }
    s = vq_block_reduce(s);
    if (threadIdx.x == 0) atomicAdd(acc, (double)s);
}

__global__ __launch_bounds__(TPB)
void vq_recon_mse(const float* __restrict__ recon, const float* __restrict__ x,
                  double* __restrict__ acc, size_t n)
{
    size_t i0 = (size_t)blockIdx.x * blockDim.x + threadIdx.x;
    size_t st = (size_t)gridDim.x * blockDim.x;
    float s = 0.f;
    for (size_t i = i0; i < n; i += st) {
        float xv = fminf(fmaxf(x[i], -10.f), 10.f);
        float d  = recon[i] - xv;
        s += d * d;
    }
    s = vq_block_reduce(s);
    if (threadIdx.x == 0) atomicAdd(acc, (double)s);
}

__global__ void vq_finalize(const double* __restrict__ acc2, float* __restrict__ losses)
{
    double emb = acc2[0] / ((double)N_ROWS * (double)DIM);
    double rec = acc2[1] / ((double)N_ROWS * (double)IN_DIM);
    losses[0] = (float)(rec + emb + 0.25 * emb);  // vq_loss
    losses[1] = (float)rec;                       // recon_loss
    losses[2] = (float)emb;                       // embedding_loss
    losses[3] = (float)emb;                       // commitment_loss (same fwd value)
}

// ---------------- launcher ----------------
extern "C" void kernel_launch(void* const* d_in, const int* in_sizes, int n_in,
                              void* d_out, int out_size, void* d_ws, size_t ws_size,
                              hipStream_t stream)
{
    const float* roi = (const float*)d_in[0];
    const float* Wp  = (const float*)d_in[1];
    const float* bp  = (const float*)d_in[2];
    const float* We  = (const float*)d_in[3];
    const float* be  = (const float*)d_in[4];
    const float* cb  = (const float*)d_in[5];
    const float* Wd  = (const float*)d_in[6];
    const float* bd  = (const float*)d_in[7];
    const float* Wo  = (const float*)d_in[8];
    const float* bo  = (const float*)d_in[9];

    char* ws = (char*)d_ws;
    float*  ze   = (float*)(ws);                                   // 64 MB
    float*  tmp  = (float*)(ws + (size_t)64  * 1024 * 1024);       // 64 MB
    float*  pval = (float*)(ws + (size_t)128 * 1024 * 1024);       // 4 MB
    int*    pidx = (int*)  (ws + (size_t)132 * 1024 * 1024);       // 4 MB
    float*  cn   = (float*)(ws + (size_t)136 * 1024 * 1024);       // 32 KB
    int*    idsI = (int*)  (ws + (size_t)136 * 1024 * 1024 + 65536);
    double* acc2 = (double*)(ws + (size_t)136 * 1024 * 1024 + 65536 + 65536);

    float* out_ids   = (float*)d_out;                              // [16384]
    float* out_zq    = out_ids + N_ROWS;                           // [16384*1024]
    float* out_recon = out_zq + (size_t)N_ROWS * DIM;              // [16384*2048]
    float* out_loss  = out_recon + (size_t)N_ROWS * IN_DIM;        // [4]
    float* out_usage = out_loss + 4;                               // [8192]

    const int nParts = CB_SIZE / 128;                              // 64

    vq_init<<<32, TPB, 0, stream>>>(acc2, out_usage);
    vq_rowsumsq<<<CB_SIZE / 8, TPB, 0, stream>>>(cb, cn, DIM);

    // z1 = clip(x) @ Wp + bp
    vq_gemm_clip<<<dim3(DIM / 128, N_ROWS / 128), TPB, 0, stream>>>(
        roi, Wp, bp, tmp, N_ROWS, DIM, IN_DIM);
    // z_e = z1 @ We + be
    vq_gemm_async<<<dim3(DIM / 128, N_ROWS / 128), TPB, 0, stream>>>(
        tmp, We, be, ze, N_ROWS, DIM, DIM);

    // nearest codebook entry
    vq_dist_partial<<<dim3(nParts, N_ROWS / 128), TPB, 0, stream>>>(
        ze, cb, cn, pval, pidx, N_ROWS, DIM);
    vq_argmin_reduce<<<N_ROWS / TPB, TPB, 0, stream>>>(
        pval, pidx, nParts, N_ROWS, idsI, out_ids);

    vq_hist<<<N_ROWS / TPB, TPB, 0, stream>>>(idsI, out_usage, N_ROWS);
    vq_gather_embloss<<<N_ROWS, TPB, 0, stream>>>(idsI, cb, ze, out_zq, &acc2[0]);

    // decoder: h = z_q @ Wd + bd ; recon = h @ Wo + bo
    vq_gemm_async<<<dim3(DIM / 128, N_ROWS / 128), TPB, 0, stream>>>(
        out_zq, Wd, bd, tmp, N_ROWS, DIM, DIM);
    vq_gemm_async<<<dim3(IN_DIM / 128, N_ROWS / 128), TPB, 0, stream>>>(
        tmp, Wo, bo, out_recon, N_ROWS, IN_DIM, DIM);

    vq_recon_mse<<<2048, TPB, 0, stream>>>(
        out_recon, roi, &acc2[1], (size_t)N_ROWS * IN_DIM);
    vq_finalize<<<1, 1, 0, stream>>>(acc2, out_loss);
}